// MultiHeadAttention_9844065042631
// MI455X (gfx1250) — compile-verified
//
#include <hip/hip_runtime.h>
#include <hip/hip_bf16.h>
#include <math.h>

// ---------------------------------------------------------------------------
// Problem constants (from reference): B=16, D_MODEL=1024, H=16, T=4096, DEPTH=64
// ---------------------------------------------------------------------------
#define BATCH  16
#define DM     1024
#define NH     16
#define TMAX   4096
#define DEPTH  64

typedef __attribute__((ext_vector_type(2))) float v2f;
typedef __attribute__((ext_vector_type(8))) float v8f;

// ---------------------------------------------------------------------------
// 16xDM @ DMxDM GEMM + bias using V_WMMA_F32_16X16X4_F32 (full fp32 precision).
// One wave (32 lanes) per 16-wide output column tile. M=16 == batch.
//
// ISA layouts (CDNA5 ISA §7.12.2):
//   A (16x4 f32): lanes 0-15 row M=lane, VGPR0=K0,VGPR1=K1; lanes 16-31 K2/K3.
//   B (4x16 f32): lanes 0-15 col N=lane, VGPR0=K0,VGPR1=K1; lanes 16-31 K2/K3.
//   C/D (16x16 f32): VGPR r -> M=r (lanes 0-15) / M=r+8 (lanes 16-31), N=lane&15.
// ---------------------------------------------------------------------------
__global__ __launch_bounds__(32) void proj16_wmma(const float* __restrict__ X,
                                                  const float* __restrict__ W,
                                                  const float* __restrict__ bias,
                                                  float* __restrict__ Y) {
    const int lane  = threadIdx.x;       // 0..31 (wave32)
    const int half  = lane >> 4;         // 0 or 1
    const int idx16 = lane & 15;         // row for A, col for B/C/D
    const int n0    = blockIdx.x * 16;   // output column tile base

    v8f acc = {};
    const float* xrow = X + (size_t)idx16 * DM;

#pragma unroll 4
    for (int k0 = 0; k0 < DM; k0 += 4) {
        const int ka = k0 + 2 * half;
        v2f a, b;
        a.x = xrow[ka];
        a.y = xrow[ka + 1];
        b.x = W[(size_t)ka * DM + n0 + idx16];
        b.y = W[(size_t)(ka + 1) * DM + n0 + idx16];
        // 8 args: (neg_a, A, neg_b, B, c_mod, C, reuse_a, reuse_b)
        acc = __builtin_amdgcn_wmma_f32_16x16x4_f32(
            false, a, false, b, (short)0, acc, false, false);
    }

    const float bv = bias[n0 + idx16];
#pragma unroll
    for (int r = 0; r < 8; ++r) {
        const int m = r + 8 * half;
        Y[(size_t)m * DM + n0 + idx16] = acc[r] + bv;
    }
}

// ---------------------------------------------------------------------------
// Fused attention + cache update/copy. One workgroup per (b,h); 256 threads.
// Streams K once (dot w/ q + write k_cache_out), exact softmax via LDS logits,
// streams V once (A.V + write v_cache_out). Row t==decode_step is replaced by
// the freshly projected kh/vh from workspace (never touches the input caches).
// ---------------------------------------------------------------------------
__global__ __launch_bounds__(256) void attn_kernel(const float* __restrict__ kc_in,
                                                   const float* __restrict__ vc_in,
                                                   const float* __restrict__ qh_ws,
                                                   const float* __restrict__ kh_ws,
                                                   const float* __restrict__ vh_ws,
                                                   const int*   __restrict__ step_p,
                                                   float* __restrict__ attn_out,
                                                   float* __restrict__ kc_out,
                                                   float* __restrict__ vc_out,
                                                   float* __restrict__ ctx_ws) {
    __shared__ float qs[DEPTH];
    __shared__ float logits_s[TMAX];
    __shared__ float red[256];

    const int tid  = threadIdx.x;
    const int b    = blockIdx.x / NH;
    const int h    = blockIdx.x % NH;
    const int step = *step_p;
    const int headoff = b * DM + h * DEPTH;   // into [B, H*DEPTH] ws tensors

    if (tid < DEPTH) qs[tid] = qh_ws[headoff + tid];
    __syncthreads();

    // ---- Phase A: logits = (K . q)/8 (+mask); copy K rows to output cache --
    float lmax = -1e30f;
#pragma unroll
    for (int i = 0; i < TMAX / 256; ++i) {
        const int t = tid + 256 * i;
        const size_t rowoff = ((size_t)t * (BATCH * NH) + b * NH + h) * DEPTH;
        const float4* src = (t == step) ? (const float4*)(kh_ws + headoff)
                                        : (const float4*)(kc_in + rowoff);
        float4* dst = (float4*)(kc_out + rowoff);
        float dot = 0.f;
#pragma unroll
        for (int j = 0; j < DEPTH / 4; ++j) {
            const float4 kv = src[j];
            dst[j] = kv;
            dot += kv.x * qs[4 * j] + kv.y * qs[4 * j + 1] +
                   kv.z * qs[4 * j + 2] + kv.w * qs[4 * j + 3];
        }
        float logit = dot * 0.125f;                 // 1/sqrt(64)
        if (t > step) logit += -1e9f;               // causal mask (matches ref)
        logits_s[t] = logit;
        lmax = fmaxf(lmax, logit);
    }

    // ---- block max reduce ----
    red[tid] = lmax;
    __syncthreads();
    for (int s = 128; s > 0; s >>= 1) {
        if (tid < s) red[tid] = fmaxf(red[tid], red[tid + s]);
        __syncthreads();
    }
    const float m = red[0];
    __syncthreads();

    // ---- Phase B: exp / sum / normalize; write attn output -----------------
    float lsum = 0.f;
#pragma unroll
    for (int i = 0; i < TMAX / 256; ++i) {
        const int t = tid + 256 * i;
        const float e = __expf(logits_s[t] - m);    // masked -> exp(-1e9) == 0
        logits_s[t] = e;
        lsum += e;
    }
    red[tid] = lsum;
    __syncthreads();
    for (int s = 128; s > 0; s >>= 1) {
        if (tid < s) red[tid] += red[tid + s];
        __syncthreads();
    }
    const float inv = 1.0f / red[0];
    __syncthreads();

    float* attn_row = attn_out + (size_t)(b * NH + h) * TMAX;
#pragma unroll
    for (int i = 0; i < TMAX / 256; ++i) {
        const int t = tid + 256 * i;
        const float a = logits_s[t] * inv;
        logits_s[t] = a;
        attn_row[t] = a;
    }
    __syncthreads();

    // ---- Phase C: ctx = attn . V; copy V rows to output cache --------------
    const int d = tid & (DEPTH - 1);
    const int g = tid >> 6;                         // 4 t-groups of 1024
    float acc = 0.f;
    for (int t = g * (TMAX / 4); t < (g + 1) * (TMAX / 4); ++t) {
        const size_t rowoff = ((size_t)t * (BATCH * NH) + b * NH + h) * DEPTH;
        const float vv = (t == step) ? vh_ws[headoff + d] : vc_in[rowoff + d];
        vc_out[rowoff + d] = vv;
        acc += logits_s[t] * vv;
    }
    red[tid] = acc;
    __syncthreads();
    if (g == 0) {
        const float c = red[d] + red[64 + d] + red[128 + d] + red[192 + d];
        ctx_ws[headoff + d] = c;
    }
}

// ---------------------------------------------------------------------------
// Launcher. Inputs (setup_inputs order):
//  0:q 1:k 2:v 3:k_cache 4:v_cache 5:Wq 6:bq 7:Wk 8:bk 9:Wv 10:bv 11:Wo 12:bo
//  13:decode_step (int, 1 elem)
// Output (concat): out[16*1024] | attn[16*16*4096] | k_cache | v_cache
// ---------------------------------------------------------------------------
extern "C" void kernel_launch(void* const* d_in, const int* in_sizes, int n_in,
                              void* d_out, int out_size, void* d_ws, size_t ws_size,
                              hipStream_t stream) {
    const float* q   = (const float*)d_in[0];
    const float* k   = (const float*)d_in[1];
    const float* v   = (const float*)d_in[2];
    const float* kc  = (const float*)d_in[3];
    const float* vc  = (const float*)d_in[4];
    const float* Wq  = (const float*)d_in[5];
    const float* bq  = (const float*)d_in[6];
    const float* Wk  = (const float*)d_in[7];
    const float* bk  = (const float*)d_in[8];
    const float* Wv  = (const float*)d_in[9];
    const float* bv  = (const float*)d_in[10];
    const float* Wo  = (const float*)d_in[11];
    const float* bo  = (const float*)d_in[12];
    const int*   stp = (const int*)d_in[13];

    float* out = (float*)d_out;
    float* ws  = (float*)d_ws;

    // d_out regions (floats)
    const size_t OFF_OUT  = 0;
    const size_t OFF_ATTN = (size_t)BATCH * DM;                       // 16384
    const size_t OFF_KC   = OFF_ATTN + (size_t)BATCH * NH * TMAX;     // +1048576
    const size_t OFF_VC   = OFF_KC + (size_t)TMAX * BATCH * NH * DEPTH;

    // workspace regions (floats): qh | kh | vh | ctx  (4 * 16K floats = 256 KB)
    float* qh  = ws;
    float* kh  = ws + (size_t)BATCH * DM;
    float* vh  = ws + (size_t)2 * BATCH * DM;
    float* ctx = ws + (size_t)3 * BATCH * DM;

    // QKV projections (fp32 WMMA, 16x16 tiles, 64 tiles each)
    proj16_wmma<<<DM / 16, 32, 0, stream>>>(q, Wq, bq, qh);
    proj16_wmma<<<DM / 16, 32, 0, stream>>>(k, Wk, bk, kh);
    proj16_wmma<<<DM / 16, 32, 0, stream>>>(v, Wv, bv, vh);

    // Fused attention + cache update/copy: one WG per (b,h)
    attn_kernel<<<BATCH * NH, 256, 0, stream>>>(
        kc, vc, qh, kh, vh, stp,
        out + OFF_ATTN, out + OFF_KC, out + OFF_VC, ctx);

    // Output projection
    proj16_wmma<<<DM / 16, 32, 0, stream>>>(ctx, Wo, bo, out + OFF_OUT);
}